// myNN_41317585388123
// MI455X (gfx1250) — compile-verified
//
#include <hip/hip_runtime.h>

typedef __attribute__((ext_vector_type(2))) float v2f;
typedef __attribute__((ext_vector_type(8))) float v8f;

#define EMB   64
#define SEQ   256
#define NBAT  16
#define NTOK  4096      /* NBAT*SEQ */
#define VOCAB 50257

__device__ __forceinline__ v8f wmma_f32(v2f a, v2f b, v8f c) {
  // D = A(16x4,f32) * B(4x16,f32) + C(16x16,f32)
  return __builtin_amdgcn_wmma_f32_16x16x4_f32(
      /*neg_a=*/false, a, /*neg_b=*/false, b,
      /*c_mod=*/(short)0, c, /*reuse_a=*/false, /*reuse_b=*/false);
}

// ---------------- 1. h = embed_w[x] + pos_w ----------------
__global__ void embed_kernel(const int* __restrict__ x,
                             const float* __restrict__ embed_w,
                             const float* __restrict__ pos_w,
                             float* __restrict__ h) {
  int i  = blockIdx.x * blockDim.x + threadIdx.x;   // over NTOK*EMB
  int bt = i >> 6;
  int e  = i & 63;
  int t  = bt & (SEQ - 1);
  int tok = x[bt];
  h[i] = embed_w[tok * EMB + e] + pos_w[t * EMB + e];
}

// ---------------- 2. Q,K,V = h @ W^T (fused) ----------------
__global__ void qkv_kernel(const float* __restrict__ h,
                           const float* __restrict__ Wq,
                           const float* __restrict__ Wk,
                           const float* __restrict__ Wv,
                           float* __restrict__ Q,
                           float* __restrict__ K,
                           float* __restrict__ V) {
  int wave = blockIdx.x * (blockDim.x >> 5) + (threadIdx.x >> 5);
  int lane = threadIdx.x & 31;
  int half = lane >> 4, lrow = lane & 15;

  int mat   = wave >> 10;            // 0..2 ; 1024 tiles per matrix
  int rem   = wave & 1023;
  int mtile = rem >> 2;              // 0..255
  int ntile = rem & 3;               // 0..3

  const float* W = (mat == 0) ? Wq : (mat == 1) ? Wk : Wv;
  float*       O = (mat == 0) ? Q  : (mat == 1) ? K  : V;

  int m0 = mtile << 4, n0 = ntile << 4;
  const float* arow = h + (m0 + lrow) * EMB + half * 2;   // A: h rows, K=e
  const float* brow = W + (n0 + lrow) * EMB + half * 2;   // B[e,f] = W[f,e]
  v8f c = {};
#pragma unroll
  for (int kk = 0; kk < EMB / 4; ++kk) {
    v2f a = *(const v2f*)(arow + kk * 4);
    v2f b = *(const v2f*)(brow + kk * 4);
    c = wmma_f32(a, b, c);
  }
  float* orow = O + (m0 + half * 8) * EMB + n0 + lrow;
#pragma unroll
  for (int r = 0; r < 8; ++r) orow[r * EMB] = c[r];
}

// ---------------- 3. att = Q K^T / 8 (skip masked tiles) ----------------
__global__ void score_kernel(const float* __restrict__ Q,
                             const float* __restrict__ K,
                             float* __restrict__ att) {
  int wave = blockIdx.x * (blockDim.x >> 5) + (threadIdx.x >> 5);
  int lane = threadIdx.x & 31;
  int half = lane >> 4, lrow = lane & 15;
  int b     = wave >> 8;             // 256 tiles per batch
  int rem   = wave & 255;
  int ttile = rem >> 4;
  int stile = rem & 15;
  if (stile > ttile) return;         // fully masked tile (wave-uniform)
  int t0 = ttile << 4, s0 = stile << 4;
  const float* qrow = Q + ((b * SEQ) + t0 + lrow) * EMB + half * 2;
  const float* krow = K + ((b * SEQ) + s0 + lrow) * EMB + half * 2; // B[e,s]=K[s,e]
  v8f c = {};
#pragma unroll
  for (int kk = 0; kk < EMB / 4; ++kk) {
    v2f a  = *(const v2f*)(qrow + kk * 4);
    v2f bb = *(const v2f*)(krow + kk * 4);
    c = wmma_f32(a, bb, c);
  }
  float* orow = att + ((size_t)b * SEQ + t0 + half * 8) * SEQ + s0 + lrow;
#pragma unroll
  for (int r = 0; r < 8; ++r) orow[r * SEQ] = c[r] * 0.125f;
}

// ---------------- 4. causal softmax over s, one block per (b,t) row ----------------
__global__ void softmax_kernel(float* __restrict__ att) {
  __shared__ float red[SEQ];
  int row = blockIdx.x;              // b*SEQ + t
  int t   = row & (SEQ - 1);
  int s   = threadIdx.x;
  float* p = att + (size_t)row * SEQ;
  float v = (s <= t) ? p[s] : -__builtin_inff();
  red[s] = v;
  __syncthreads();
  for (int off = 128; off > 0; off >>= 1) {
    if (s < off) red[s] = fmaxf(red[s], red[s + off]);
    __syncthreads();
  }
  float m = red[0];
  __syncthreads();
  float ex = (s <= t) ? __expf(v - m) : 0.f;
  red[s] = ex;
  __syncthreads();
  for (int off = 128; off > 0; off >>= 1) {
    if (s < off) red[s] += red[s + off];
    __syncthreads();
  }
  p[s] = ex * (1.f / red[0]);
}

// ---------------- 5. ctx = att @ V ----------------
__global__ void ctx_kernel(const float* __restrict__ att,
                           const float* __restrict__ V,
                           float* __restrict__ ctx) {
  int wave = blockIdx.x * (blockDim.x >> 5) + (threadIdx.x >> 5);
  int lane = threadIdx.x & 31;
  int half = lane >> 4, lrow = lane & 15;
  int b    = wave >> 6;              // 64 tiles per batch
  int rem  = wave & 63;
  int ttile = rem >> 2;
  int ftile = rem & 3;
  int t0 = ttile << 4, f0 = ftile << 4;
  const float* arow = att + ((size_t)b * SEQ + t0 + lrow) * SEQ + half * 2;
  const float* vb   = V + (size_t)b * SEQ * EMB;
  v8f c = {};
  int kmax = (t0 + 16) >> 2;         // att is 0 beyond the causal edge
  for (int kk = 0; kk < kmax; ++kk) {
    int s = kk * 4 + half * 2;
    v2f a = *(const v2f*)(arow + kk * 4);
    v2f bb;
    bb.x = vb[(s + 0) * EMB + f0 + lrow];   // B[s,f] = V[s,f]
    bb.y = vb[(s + 1) * EMB + f0 + lrow];
    c = wmma_f32(a, bb, c);
  }
  float* orow = ctx + ((size_t)b * SEQ + t0 + half * 8) * EMB + f0 + lrow;
#pragma unroll
  for (int r = 0; r < 8; ++r) orow[r * EMB] = c[r];
}

// ---------------- 6. logits = ctx @ fc_w^T + fc_b ----------------
// 2 m-tiles x 4 n-tiles (32x64 output) per wave: A reuse 4x, B reuse 2x,
// L2 read traffic ~2.4 GB vs 823 MB of NT stores -> store-bound as intended.
__global__ void logits_kernel(const float* __restrict__ ctx,
                              const float* __restrict__ fc_w,
                              const float* __restrict__ fc_b,
                              float* __restrict__ out) {
  const int NBLKN = (VOCAB + 63) / 64;     // 786 n-blocks of 64 cols
  int wave = blockIdx.x * (blockDim.x >> 5) + (threadIdx.x >> 5);
  int lane = threadIdx.x & 31;
  int half = lane >> 4, lrow = lane & 15;
  int mblk = wave / NBLKN;                 // 0..127 (32 rows each)
  int nblk = wave - mblk * NBLKN;
  if (mblk >= NTOK / 32) return;           // wave-uniform
  int m0 = mblk * 32;
  int n0 = nblk * 64;

  const float* arow0 = ctx + (m0 +      lrow) * EMB + half * 2;
  const float* arow1 = ctx + (m0 + 16 + lrow) * EMB + half * 2;

  int  vcol[4];
  bool valid[4];
  const float* brow[4];
#pragma unroll
  for (int j = 0; j < 4; ++j) {
    vcol[j]  = n0 + j * 16 + lrow;
    valid[j] = vcol[j] < VOCAB;
    brow[j]  = fc_w + (size_t)vcol[j] * EMB + half * 2;  // B[e,v] = fc_w[v,e]
  }

  v8f acc[2][4] = {};
  const v2f zero = {0.f, 0.f};
#pragma unroll
  for (int kk = 0; kk < EMB / 4; ++kk) {
    v2f a0 = *(const v2f*)(arow0 + kk * 4);
    v2f a1 = *(const v2f*)(arow1 + kk * 4);
    v2f b[4];
#pragma unroll
    for (int j = 0; j < 4; ++j)
      b[j] = valid[j] ? *(const v2f*)(brow[j] + kk * 4) : zero;
#pragma unroll
    for (int j = 0; j < 4; ++j) {
      acc[0][j] = wmma_f32(a0, b[j], acc[0][j]);   // EXEC full here
      acc[1][j] = wmma_f32(a1, b[j], acc[1][j]);
    }
  }

#pragma unroll
  for (int j = 0; j < 4; ++j) {
    if (valid[j]) {
      float bias = fc_b[vcol[j]];
#pragma unroll
      for (int i = 0; i < 2; ++i) {
        float* orow = out + (size_t)(m0 + i * 16 + half * 8) * VOCAB + vcol[j];
#pragma unroll
        for (int r = 0; r < 8; ++r)
          __builtin_nontemporal_store(acc[i][j][r] + bias, orow + (size_t)r * VOCAB);
      }
    }
  }
}

extern "C" void kernel_launch(void* const* d_in, const int* in_sizes, int n_in,
                              void* d_out, int out_size, void* d_ws, size_t ws_size,
                              hipStream_t stream) {
  const int*   x       = (const int*)d_in[0];
  const float* embed_w = (const float*)d_in[1];
  const float* pos_w   = (const float*)d_in[2];
  const float* Wq      = (const float*)d_in[3];
  const float* Wk      = (const float*)d_in[4];
  const float* Wv      = (const float*)d_in[5];
  const float* fc_w    = (const float*)d_in[6];
  const float* fc_b    = (const float*)d_in[7];
  float* out = (float*)d_out;

  float* ws  = (float*)d_ws;
  float* h   = ws;                   // 262144 f32
  float* Q   = ws + 262144;          // 262144
  float* K   = ws + 524288;          // 262144
  float* V   = ws + 786432;          // 262144
  float* att = ws + 1048576;         // 1048576
  float* ctx = ws + 2097152;         // 262144   (total ~9 MB)

  embed_kernel  <<<NTOK * EMB / 256, 256, 0, stream>>>(x, embed_w, pos_w, h);
  qkv_kernel    <<<3 * 1024 / 8,     256, 0, stream>>>(h, Wq, Wk, Wv, Q, K, V);
  score_kernel  <<<NBAT * 256 / 8,   256, 0, stream>>>(Q, K, att);
  softmax_kernel<<<NTOK,             256, 0, stream>>>(att);
  ctx_kernel    <<<NBAT * 64 / 8,    256, 0, stream>>>(att, V, ctx);

  const int NBLKN = (VOCAB + 63) / 64;           // 786
  int waves  = (NTOK / 32) * NBLKN;              // 100608
  logits_kernel <<<(waves + 7) / 8,  256, 0, stream>>>(ctx, fc_w, fc_b, out);
}